// Attention_22754736734691
// MI455X (gfx1250) — compile-verified
//
#include <hip/hip_runtime.h>
#include <math.h>

// ---------------------------------------------------------------------------
// Paged-attention decode for MI455X (gfx1250, wave32).
// Flash-decoding split-K: 8 KV heads x 128 partitions of 128 tokens each.
// QK^T runs on the matrix pipe via V_WMMA_F32_16X16X4_F32 (A = 4 grouped Q
// heads padded to 16 rows, B = 16 keys, 32 K-steps cover head_dim=128).
// Memory bound (~2.1 FLOP/B): design maximizes coalesced fp32 streaming.
// ---------------------------------------------------------------------------

typedef float v2f __attribute__((ext_vector_type(2)));
typedef float v4f __attribute__((ext_vector_type(4)));
typedef float v8f __attribute__((ext_vector_type(8)));

#define HQ     32
#define HKV    8
#define GRP    4         // HQ / HKV
#define DH     128
#define BS     16        // cache block size == WMMA N tile
#define NPART  128       // context partitions per kv head
#define PART   128       // tokens per partition (one wave each)
#define TILES  (PART / BS)
#define WAVES  8         // waves per workgroup
#define SCALE  0.08838834764831845f
#define NEG_BIG (-3.0e38f)

__global__ __launch_bounds__(256) void pa_phase1(
    const float* __restrict__ q, const float* __restrict__ knew,
    const float* __restrict__ vnew,
    const float* __restrict__ kc, const float* __restrict__ vc,
    const int* __restrict__ slot_mapping, const int* __restrict__ block_table,
    const int* __restrict__ Tptr,
    float* __restrict__ accW, float* __restrict__ mlW)
{
    const int g    = blockIdx.x;               // kv head
    const int wave = threadIdx.x >> 5;
    const int lane = threadIdx.x & 31;
    const int half = lane >> 4;                // hi/lo half-wave (K-dim pairing)
    const int mrow = lane & 15;                // WMMA row (M) / key index (N)
    const int part = blockIdx.y * WAVES + wave;
    const int t0   = part * PART;
    const int T    = Tptr[0];
    const int slot_new = slot_mapping[0];

    // --- Q fragments for all 32 WMMA K-steps (A-matrix layout, stays in VGPRs)
    float aq0[32], aq1[32];
    const bool qvalid = (mrow < GRP);
    const float* qrow = q + (g * GRP + (qvalid ? mrow : 0)) * DH;
    #pragma unroll
    for (int ks = 0; ks < 32; ++ks) {
        const int d = 4 * ks + 2 * half;
        const float x0 = qrow[d];
        const float x1 = qrow[d + 1];
        aq0[ks] = qvalid ? x0 : 0.0f;
        aq1[ks] = qvalid ? x1 : 0.0f;
    }

    float mrun[GRP], lrun[GRP], acc[GRP][4];
    #pragma unroll
    for (int i = 0; i < GRP; ++i) {
        mrun[i] = NEG_BIG; lrun[i] = 0.0f;
        #pragma unroll
        for (int j = 0; j < 4; ++j) acc[i][j] = 0.0f;
    }

    for (int tt = 0; tt < TILES; ++tt) {
        const int tb = t0 + tt * BS;           // tile aligned with a cache block
        const int pb = block_table[tb >> 4];   // physical block id
        const int slot_l = pb * BS + mrow;     // this lane's key slot

        // new-token merge: reference writes k/v into cache at slot_mapping[0]
        const float* kptr = (slot_l == slot_new)
                          ? (knew + g * DH)
                          : (kc + ((size_t)slot_l * HKV + g) * DH);
        const float* vrow_pf = vc + ((size_t)slot_l * HKV + g) * DH;
        __builtin_prefetch(vrow_pf + 4 * lane, 0, 1);   // warm V while QK runs

        // ---- scores = Q x K^T over 32 WMMA steps (EXEC all-ones here) ----
        v8f c = {};
        #pragma unroll
        for (int ks = 0; ks < 32; ++ks) {
            v2f a; a.x = aq0[ks]; a.y = aq1[ks];
            v2f b = *(const v2f*)(kptr + 4 * ks + 2 * half);
            c = __builtin_amdgcn_wmma_f32_16x16x4_f32(
                    false, a, false, b, (short)0, c, false, false);
        }

        // ---- online softmax per grouped q-head (rows 0..3, lanes 0..15) ----
        float p[GRP];
        #pragma unroll
        for (int mm = 0; mm < GRP; ++mm) {
            float s = c[mm] * SCALE;
            const bool dead = (tb + mrow >= T);
            if (dead) s = NEG_BIG;
            float mx = s;
            mx = fmaxf(mx, __shfl_xor(mx, 8, 16));
            mx = fmaxf(mx, __shfl_xor(mx, 4, 16));
            mx = fmaxf(mx, __shfl_xor(mx, 2, 16));
            mx = fmaxf(mx, __shfl_xor(mx, 1, 16));
            mx = __shfl(mx, 0, 32);            // wave-uniform row max
            const float mnew = fmaxf(mrun[mm], mx);
            float pe = __expf(s - mnew);
            if (dead) pe = 0.0f;
            float sum = pe;
            sum += __shfl_xor(sum, 8, 16);
            sum += __shfl_xor(sum, 4, 16);
            sum += __shfl_xor(sum, 2, 16);
            sum += __shfl_xor(sum, 1, 16);
            sum = __shfl(sum, 0, 32);          // wave-uniform row sum
            const float corr = __expf(mrun[mm] - mnew);
            lrun[mm] = lrun[mm] * corr + sum;
            mrun[mm] = mnew;
            #pragma unroll
            for (int j = 0; j < 4; ++j) acc[mm][j] *= corr;
            p[mm] = pe;                        // valid in lanes 0..15 (lane = key)
        }

        // ---- acc += P x V : b128 coalesced V rows, readlane P broadcast ----
        #pragma unroll
        for (int n = 0; n < BS; ++n) {
            const int slot_n = pb * BS + n;
            const float* vptr = (slot_n == slot_new)
                              ? (vnew + g * DH)
                              : (vc + ((size_t)slot_n * HKV + g) * DH);
            const v4f vv = *(const v4f*)(vptr + 4 * lane);
            #pragma unroll
            for (int mm = 0; mm < GRP; ++mm) {
                const float pn = __shfl(p[mm], n, 32);
                acc[mm][0] += pn * vv.x;
                acc[mm][1] += pn * vv.y;
                acc[mm][2] += pn * vv.z;
                acc[mm][3] += pn * vv.w;
            }
        }
    }

    // ---- write partition partials: unnormalized acc + (m, l) ----
    #pragma unroll
    for (int mm = 0; mm < GRP; ++mm) {
        const int qh = g * GRP + mm;
        float* dst = accW + ((size_t)(qh * NPART + part)) * DH + 4 * lane;
        v4f o; o.x = acc[mm][0]; o.y = acc[mm][1]; o.z = acc[mm][2]; o.w = acc[mm][3];
        *(v4f*)dst = o;
        if (lane == 0) {
            mlW[(qh * NPART + part) * 2 + 0] = mrun[mm];
            mlW[(qh * NPART + part) * 2 + 1] = lrun[mm];
        }
    }
}

__global__ __launch_bounds__(128) void pa_phase2(
    const float* __restrict__ accW, const float* __restrict__ mlW,
    float* __restrict__ out)
{
    const int qh = blockIdx.x;   // 0..31
    const int d  = threadIdx.x;  // 0..127
    float M = NEG_BIG, num = 0.0f, den = 0.0f;
    for (int p = 0; p < NPART; ++p) {
        const float mp = mlW[(qh * NPART + p) * 2 + 0];
        const float lp = mlW[(qh * NPART + p) * 2 + 1];
        const float ap = accW[((size_t)(qh * NPART + p)) * DH + d];
        const float Mn    = fmaxf(M, mp);
        const float alpha = __expf(M - Mn);
        const float beta  = __expf(mp - Mn);
        num = num * alpha + ap * beta;
        den = den * alpha + lp * beta;
        M = Mn;
    }
    out[qh * DH + d] = num / den;   // output layout (1,1,32,128) == [qh][d]
}

extern "C" void kernel_launch(void* const* d_in, const int* in_sizes, int n_in,
                              void* d_out, int out_size, void* d_ws, size_t ws_size,
                              hipStream_t stream)
{
    const float* q    = (const float*)d_in[0];
    const float* k    = (const float*)d_in[1];
    const float* v    = (const float*)d_in[2];
    const float* kc   = (const float*)d_in[3];
    const float* vc   = (const float*)d_in[4];
    const int*   slot = (const int*)d_in[5];
    const int*   btab = (const int*)d_in[6];
    const int*   Tp   = (const int*)d_in[7];

    float* accW = (float*)d_ws;                       // 32*128*128 floats = 2 MB
    float* mlW  = accW + (size_t)HQ * NPART * DH;     // 32*128*2 floats

    dim3 grid1(HKV, NPART / WAVES);                   // (8, 16) blocks x 8 waves
    pa_phase1<<<grid1, 256, 0, stream>>>(q, k, v, kc, vc, slot, btab, Tp,
                                         accW, mlW);
    pa_phase2<<<HQ, DH, 0, stream>>>(accW, mlW, (float*)d_out);
}